// StressDeepGCN_28724741275673
// MI455X (gfx1250) — compile-verified
//
#include <hip/hip_runtime.h>
#include <hip/hip_bf16.h>

#define N_NODES 100000
#define N_EDGES 1200000
#define CIN_DIM 3
#define H 64
#define H2 128
#define OUT_DIM 3
#define MSG_EPS 1e-7f
#define LN_EPS 1e-5f
#define TILES_PER_BLOCK 2
#define PACKED_PER_LAYER (2 * H * H2)   /* W1(8192) + W2(8192) floats */

typedef __attribute__((ext_vector_type(2))) float v2f;
typedef __attribute__((ext_vector_type(8))) float v8f;

// ---- monotonic f32 <-> u32 encoding so atomicMax(u32) == float max ----
__device__ __forceinline__ unsigned enc_f32(float f) {
  unsigned u = __float_as_uint(f);
  return (u & 0x80000000u) ? ~u : (u | 0x80000000u);
}
__device__ __forceinline__ float dec_f32(unsigned u) {
  unsigned v = (u & 0x80000000u) ? (u & 0x7FFFFFFFu) : ~u;
  return __uint_as_float(v);
}

// ---- gfx1250 async global->LDS staging (ASYNCcnt) ----------------------
// Issues 32 x B128 per thread (64-thread block stages 32KB), no wait here.
__device__ __forceinline__ void async_stage_32kb_issue(const float* src,
                                                       float* dst_lds,
                                                       int tid) {
  unsigned lbase = (unsigned)(unsigned long long)(void*)dst_lds;
#pragma unroll
  for (int it = 0; it < 32; ++it) {
    unsigned goff = (unsigned)(tid + it * 64) * 16u;
    unsigned laddr = lbase + goff;
    asm volatile("global_load_async_to_lds_b128 %0, %1, %2"
                 :: "v"(laddr), "v"(goff), "s"(src)
                 : "memory");
  }
}
__device__ __forceinline__ void async_stage_wait() {
  asm volatile("s_wait_asynccnt 0x0" ::: "memory");
}

// ==== one-shot weight repack into WMMA-fragment-native order ============
// Packed: for each 4-wide K block kb, col, half p in {0,1}:
//   wp[((kb*NC + col)*2 + p)*2 + q] = W[kb*4 + 2p + q][col]   (q in {0,1})
// so lane's B operand is a single contiguous 8B LDS load.
__global__ void pack_weights_kernel(const float* __restrict__ W1,
                                    const float* __restrict__ W2,
                                    float* __restrict__ wp) {
  int gid = blockIdx.x * blockDim.x + threadIdx.x;
  if (gid >= 3 * PACKED_PER_LAYER) return;
  int l = gid / PACKED_PER_LAYER;
  int r = gid - l * PACKED_PER_LAYER;
  float val;
  if (r < H * H2) {                       // W1: K=64 rows x 128 cols
    int i = r;
    int q = i & 1, p = (i >> 1) & 1;
    int col = (i >> 2) & (H2 - 1);
    int kb = i >> 9;
    val = W1[(size_t)l * H * H2 + (kb * 4 + 2 * p + q) * H2 + col];
  } else {                                // W2: K=128 rows x 64 cols
    int i = r - H * H2;
    int q = i & 1, p = (i >> 1) & 1;
    int col = (i >> 2) & (H - 1);
    int kb = i >> 8;
    val = W2[(size_t)l * H2 * H + (kb * 4 + 2 * p + q) * H + col];
  }
  wp[gid] = val;
}

// ===================== encoder: z = x @ enc_W + enc_b ; init tables =====
__global__ void encoder_init_kernel(const float* __restrict__ x,
                                    const float* __restrict__ W,
                                    const float* __restrict__ b,
                                    float* __restrict__ z,
                                    unsigned* __restrict__ m_u,
                                    float* __restrict__ denom,
                                    float* __restrict__ num) {
  int gid = blockIdx.x * blockDim.x + threadIdx.x;
  if (gid >= N_NODES * H) return;
  int n = gid >> 6, c = gid & 63;
  float v = b[c];
  v += x[n * CIN_DIM + 0] * W[0 * H + c];
  v += x[n * CIN_DIM + 1] * W[1 * H + c];
  v += x[n * CIN_DIM + 2] * W[2 * H + c];
  z[gid] = v;
  m_u[gid] = 0u;        // encoded -inf floor
  denom[gid] = 0.f;
  num[gid] = 0.f;
}

// ===================== edge pass 1: segment max of logits ===============
__global__ void edge_max_kernel(const float* __restrict__ z,
                                const int* __restrict__ ei,
                                const float* __restrict__ t_arr, int layer,
                                unsigned* __restrict__ m_u) {
  int gid = blockIdx.x * blockDim.x + threadIdx.x;
  if (gid >= N_EDGES * 16) return;
  int e = gid >> 4;
  int c = (gid & 15) * 4;
  int src = ei[e];
  int dst = ei[N_EDGES + e];
  float t = t_arr[layer];
  float4 v = *(const float4*)(z + src * H + c);
  int ob = dst * H + c;
  float comps[4] = {v.x, v.y, v.z, v.w};
#pragma unroll
  for (int k = 0; k < 4; k++) {
    float msg = fmaxf(comps[k], 0.f) + MSG_EPS;
    atomicMax(&m_u[ob + k], enc_f32(msg * t));
  }
}

// ===================== edge pass 2: w=exp(l-m); sum w, sum msg*w ========
__global__ void edge_wsum_kernel(const float* __restrict__ z,
                                 const int* __restrict__ ei,
                                 const float* __restrict__ t_arr, int layer,
                                 const unsigned* __restrict__ m_u,
                                 float* __restrict__ denom,
                                 float* __restrict__ num) {
  int gid = blockIdx.x * blockDim.x + threadIdx.x;
  if (gid >= N_EDGES * 16) return;
  int e = gid >> 4;
  int c = (gid & 15) * 4;
  int src = ei[e];
  int dst = ei[N_EDGES + e];
  float t = t_arr[layer];
  float4 v = *(const float4*)(z + src * H + c);
  int ob = dst * H + c;
  float comps[4] = {v.x, v.y, v.z, v.w};
#pragma unroll
  for (int k = 0; k < 4; k++) {
    float msg = fmaxf(comps[k], 0.f) + MSG_EPS;
    float m = dec_f32(m_u[ob + k]);
    float w = __expf(msg * t - m);
    atomicAdd(&denom[ob + k], w);
    atomicAdd(&num[ob + k], msg * w);
  }
}

// ===== pre-LN for layers 1,2: z = relu(LN(h)); re-init scatter tables ===
__global__ void preln_init_kernel(const float* __restrict__ h,
                                  const float* __restrict__ g,
                                  const float* __restrict__ b,
                                  float* __restrict__ z,
                                  unsigned* __restrict__ m_u,
                                  float* __restrict__ denom,
                                  float* __restrict__ num) {
  int tid = threadIdx.x;
  int lane16 = tid & 15;
  int node = blockIdx.x * 16 + (tid >> 4);   // grid = N/16 exact, all lanes live
  int base = node * H + lane16 * 4;
  float4 v = *(const float4*)(h + base);
  float s = v.x + v.y + v.z + v.w;
  float q = v.x * v.x + v.y * v.y + v.z * v.z + v.w * v.w;
#pragma unroll
  for (int m = 8; m >= 1; m >>= 1) {
    s += __shfl_xor(s, m, 32);
    q += __shfl_xor(q, m, 32);
  }
  float mu = s * (1.f / H);
  float rs = rsqrtf(q * (1.f / H) - mu * mu + LN_EPS);
  int c = lane16 * 4;
  float4 o;
  o.x = fmaxf((v.x - mu) * rs * g[c + 0] + b[c + 0], 0.f);
  o.y = fmaxf((v.y - mu) * rs * g[c + 1] + b[c + 1], 0.f);
  o.z = fmaxf((v.z - mu) * rs * g[c + 2] + b[c + 2], 0.f);
  o.w = fmaxf((v.w - mu) * rs * g[c + 3] + b[c + 3], 0.f);
  *(float4*)(z + base) = o;
  float4 zz = {0.f, 0.f, 0.f, 0.f};
  *(float4*)(denom + base) = zz;
  *(float4*)(num + base) = zz;
  uint4 uz = {0u, 0u, 0u, 0u};
  *(uint4*)(m_u + base) = uz;
}

// ===================== node MLP with f32 WMMA ===========================
// one wave32 per 16-node tile; 2 tiles (2 waves) per block of 64 threads.
// Packed weights staged block-wide into LDS with async global->LDS DMA,
// overlapped with the aggregation build.
__global__ void __launch_bounds__(64)
node_mlp_kernel(const float* __restrict__ z,
                const float* __restrict__ numb,
                const float* __restrict__ denb,
                const float* __restrict__ wp1, const float* __restrict__ b1,
                const float* __restrict__ g,   const float* __restrict__ beta,
                const float* __restrict__ wp2, const float* __restrict__ b2,
                float* __restrict__ h, int accumulate) {
  __shared__ __align__(16) float wbuf[H * H2];         // 32KB: W1 then W2
  __shared__ __align__(16) float lds_out[TILES_PER_BLOCK][16 * H];   // 2x4KB
  __shared__ __align__(16) float lds_y1[TILES_PER_BLOCK][16 * H2];   // 2x8KB
  int tid = threadIdx.x;
  int w = tid >> 5;
  int lane = tid & 31;
  int nodeBase = (blockIdx.x * TILES_PER_BLOCK + w) * 16;  // grid exact

  // Kick off async DMA of packed W1 (32KB) into LDS.
  async_stage_32kb_issue(wp1, wbuf, tid);

  // Phase 1 (overlapped with DMA): out = agg + x into LDS tile [16 x 64]
  for (int i = lane; i < 16 * H; i += 32) {
    int idx = (nodeBase + (i >> 6)) * H + (i & 63);
    float d = denb[idx];
    float a = d > 0.f ? numb[idx] / d : 0.f;
    lds_out[w][i] = a + z[idx];
  }
  async_stage_wait();       // this wave's DMA chunk is in LDS
  __syncthreads();          // all waves' chunks + out-tiles visible

  int row = lane & 15;                    // A-fragment row (M)
  int halfsel = (lane < 16) ? 0 : 1;      // K sub-pair per half-wave
  int colsel = lane & 15;                 // B/C column within 16-wide tile

  // GEMM1: y1[16x128] = out[16x64] @ W1[64x128], K in steps of 4
  v8f zero8 = {};
  v8f acc[8];
#pragma unroll
  for (int j = 0; j < 8; j++) acc[j] = zero8;
#pragma unroll
  for (int kb = 0; kb < 16; kb++) {
    v2f a = *(const v2f*)&lds_out[w][row * H + kb * 4 + 2 * halfsel];
#pragma unroll
    for (int j = 0; j < 8; j++) {
      int col = j * 16 + colsel;
      v2f bv = *(const v2f*)&wbuf[((kb * H2 + col) * 2 + halfsel) * 2];
      acc[j] = __builtin_amdgcn_wmma_f32_16x16x4_f32(
          false, a, false, bv, (short)0, acc[j], false, false);
    }
  }

  // bias + LayerNorm over 128 cols (per row) + relu -> lds_y1
  // C/D layout: element (j,v) is row M = v + 8*(lane>=16), col = 16*j+colsel
  float mu[8], rs[8];
#pragma unroll
  for (int v = 0; v < 8; v++) {
    float s = 0.f, q = 0.f;
#pragma unroll
    for (int j = 0; j < 8; j++) {
      float xv = acc[j][v] + b1[j * 16 + colsel];
      acc[j][v] = xv;
      s += xv; q += xv * xv;
    }
#pragma unroll
    for (int m = 8; m >= 1; m >>= 1) {      // xor masks <16 stay in half-wave
      s += __shfl_xor(s, m, 32);
      q += __shfl_xor(q, m, 32);
    }
    float mm = s * (1.f / H2);
    mu[v] = mm;
    rs[v] = rsqrtf(q * (1.f / H2) - mm * mm + LN_EPS);
  }
  int rowbase = (lane < 16) ? 0 : 8;
#pragma unroll
  for (int j = 0; j < 8; j++) {
    int col = j * 16 + colsel;
    float gg = g[col], bb = beta[col];
#pragma unroll
    for (int v = 0; v < 8; v++) {
      float y = fmaxf((acc[j][v] - mu[v]) * rs[v] * gg + bb, 0.f);
      lds_y1[w][(rowbase + v) * H2 + col] = y;
    }
  }
  __syncthreads();          // all waves done reading W1 from wbuf

  // Stage packed W2 (32KB) into the same LDS buffer.
  async_stage_32kb_issue(wp2, wbuf, tid);
  async_stage_wait();
  __syncthreads();

  // GEMM2: y2[16x64] = y1[16x128] @ W2[128x64]
  v8f acc2[4];
#pragma unroll
  for (int j = 0; j < 4; j++) acc2[j] = zero8;
#pragma unroll
  for (int kb = 0; kb < 32; kb++) {
    v2f a = *(const v2f*)&lds_y1[w][row * H2 + kb * 4 + 2 * halfsel];
#pragma unroll
    for (int j = 0; j < 4; j++) {
      int col = j * 16 + colsel;
      v2f bv = *(const v2f*)&wbuf[((kb * H + col) * 2 + halfsel) * 2];
      acc2[j] = __builtin_amdgcn_wmma_f32_16x16x4_f32(
          false, a, false, bv, (short)0, acc2[j], false, false);
    }
  }
  // bias + (residual) store
#pragma unroll
  for (int j = 0; j < 4; j++) {
    int col = j * 16 + colsel;
    float bb = b2[col];
#pragma unroll
    for (int v = 0; v < 8; v++) {
      int idx = (nodeBase + rowbase + v) * H + col;
      float val = acc2[j][v] + bb;
      if (accumulate) h[idx] += val; else h[idx] = val;
    }
  }
}

// ===================== head: out = relu(LN(h, ln0)) @ lin_W + lin_b =====
__global__ void head_kernel(const float* __restrict__ h,
                            const float* __restrict__ g,
                            const float* __restrict__ b,
                            const float* __restrict__ lw,
                            const float* __restrict__ lb,
                            float* __restrict__ out) {
  int tid = threadIdx.x;
  int lane16 = tid & 15;
  int node = blockIdx.x * 16 + (tid >> 4);   // grid = N/16 exact
  float4 v = *(const float4*)(h + node * H + lane16 * 4);
  float s = v.x + v.y + v.z + v.w;
  float q = v.x * v.x + v.y * v.y + v.z * v.z + v.w * v.w;
#pragma unroll
  for (int m = 8; m >= 1; m >>= 1) {
    s += __shfl_xor(s, m, 32);
    q += __shfl_xor(q, m, 32);
  }
  float mu = s * (1.f / H);
  float rs = rsqrtf(q * (1.f / H) - mu * mu + LN_EPS);
  int c = lane16 * 4;
  float comps[4] = {v.x, v.y, v.z, v.w};
  float p0 = 0.f, p1 = 0.f, p2 = 0.f;
#pragma unroll
  for (int k = 0; k < 4; k++) {
    float y = fmaxf((comps[k] - mu) * rs * g[c + k] + b[c + k], 0.f);
    p0 += y * lw[(c + k) * OUT_DIM + 0];
    p1 += y * lw[(c + k) * OUT_DIM + 1];
    p2 += y * lw[(c + k) * OUT_DIM + 2];
  }
#pragma unroll
  for (int m = 8; m >= 1; m >>= 1) {
    p0 += __shfl_xor(p0, m, 32);
    p1 += __shfl_xor(p1, m, 32);
    p2 += __shfl_xor(p2, m, 32);
  }
  if (lane16 == 0) {
    out[node * OUT_DIM + 0] = p0 + lb[0];
    out[node * OUT_DIM + 1] = p1 + lb[1];
    out[node * OUT_DIM + 2] = p2 + lb[2];
  }
}

extern "C" void kernel_launch(void* const* d_in, const int* in_sizes, int n_in,
                              void* d_out, int out_size, void* d_ws, size_t ws_size,
                              hipStream_t stream) {
  const float* x     = (const float*)d_in[0];
  const int*   ei    = (const int*)d_in[1];
  const float* enc_W = (const float*)d_in[2];
  const float* enc_b = (const float*)d_in[3];
  const float* t     = (const float*)d_in[4];
  const float* W1    = (const float*)d_in[5];
  const float* b1    = (const float*)d_in[6];
  const float* mg    = (const float*)d_in[7];
  const float* mbeta = (const float*)d_in[8];
  const float* W2    = (const float*)d_in[9];
  const float* b2    = (const float*)d_in[10];
  const float* ln_g  = (const float*)d_in[11];
  const float* ln_b  = (const float*)d_in[12];
  const float* lw    = (const float*)d_in[13];
  const float* lb    = (const float*)d_in[14];
  float* out = (float*)d_out;

  size_t NH = (size_t)N_NODES * H;
  float*    ws    = (float*)d_ws;
  float*    hbuf  = ws;                      // [N,H] layer state
  float*    zbuf  = ws + NH;                 // [N,H] conv input
  unsigned* m_u   = (unsigned*)(ws + 2 * NH);// [N,H] encoded seg-max
  float*    denom = ws + 3 * NH;             // [N,H]
  float*    numb  = ws + 4 * NH;             // [N,H]
  float*    wpack = ws + 5 * NH;             // [3][16384] packed W1|W2

  int gridNH = (N_NODES * H + 255) / 256;
  int gridE  = (N_EDGES * 16 + 255) / 256;
  int gridN16 = N_NODES / 16;                         // 6250 (exact)
  int gridMLP = gridN16 / TILES_PER_BLOCK;            // 3125 (exact)
  int gridPK  = (3 * PACKED_PER_LAYER + 255) / 256;

  pack_weights_kernel<<<gridPK, 256, 0, stream>>>(W1, W2, wpack);
  encoder_init_kernel<<<gridNH, 256, 0, stream>>>(x, enc_W, enc_b, zbuf, m_u, denom, numb);

  for (int l = 0; l < 3; l++) {
    if (l > 0) {
      preln_init_kernel<<<gridN16, 256, 0, stream>>>(hbuf, ln_g + l * H, ln_b + l * H,
                                                     zbuf, m_u, denom, numb);
    }
    edge_max_kernel <<<gridE, 256, 0, stream>>>(zbuf, ei, t, l, m_u);
    edge_wsum_kernel<<<gridE, 256, 0, stream>>>(zbuf, ei, t, l, m_u, denom, numb);
    const float* wp_l = wpack + (size_t)l * PACKED_PER_LAYER;
    node_mlp_kernel<<<gridMLP, 64, 0, stream>>>(
        zbuf, numb, denom,
        wp_l,              b1 + (size_t)l * H2,
        mg + (size_t)l * H2, mbeta + (size_t)l * H2,
        wp_l + H * H2,     b2 + (size_t)l * H,
        hbuf, l > 0 ? 1 : 0);
  }

  head_kernel<<<gridN16, 256, 0, stream>>>(hbuf, ln_g, ln_b, lw, lb, out);
}